// AKLLoss_56607668961846
// MI455X (gfx1250) — compile-verified
//
#include <hip/hip_runtime.h>

// AKL distillation loss for MI455X (gfx1250, wave32, 320KB-LDS WGP).
// One 1024-thread workgroup per (b,l) row. Both logit rows are staged into LDS
// (250KB — only possible on CDNA5) by the Tensor Data Mover: wave 0 DMAs the
// teacher row, wave 1 the student row (concurrent descriptors), each waiting on
// its own TENSORcnt. The reference's full sort is replaced by a 2-level
// histogram select (exact, tie-aware) done entirely out of LDS.

#define NT      1024
#define NWAVES  (NT / 32)
#define NB      1024          // histogram bins (== NT so scan is 1 elem/thread)
#define CAP     32            // max candidates for exact final selection
#define INV_T   0.5f          // 1/TEMPERATURE (exact power of two -> exact mul)
#define T2      4.0f          // TEMPERATURE^2
#define MU      0.5f
#define AKL_EPS 1e-8f

typedef unsigned int u32x4 __attribute__((ext_vector_type(4)));
typedef unsigned int u32x8 __attribute__((ext_vector_type(8)));

// ---- Tensor Data Mover: 1-D row DMA global -> LDS (D# per CDNA5 ISA).
// rowBytes expressed as (rowBytes/8) 8-byte elements; caller guarantees
// rowBytes % 8 == 0 and rowBytes/8 <= 0xFFFF (tile_dim0 is 16-bit).
__device__ __forceinline__ void tdm_load_row(const void* gsrc, void* ldst,
                                             unsigned rowBytes) {
  const unsigned n8 = rowBytes >> 3;
  const unsigned long long ga = (unsigned long long)(size_t)gsrc;
  const unsigned lds32 = (unsigned)(size_t)ldst;   // low 32 bits = LDS byte offset
  u32x4 g0;
  g0[0] = 1u;                                      // count=1, user mode, no gather
  g0[1] = lds32;                                   // lds_addr
  g0[2] = (unsigned)ga;                            // global_addr[31:0]
  g0[3] = (unsigned)((ga >> 32) & 0x01FFFFFFu)     // global_addr[56:32]
        | 0x80000000u;                             // type=2 ("image")
  u32x8 g1;
  g1[0] = 0x00030000u;                             // wg_mask=0, data_size=3 (8B)
  g1[1] = (n8 & 0xFFFFu) << 16;                    // tensor_dim0[15:0]
  g1[2] = (n8 >> 16) | (1u << 16);                 // tensor_dim0[31:16] | tensor_dim1=1
  g1[3] = (n8 & 0xFFFFu) << 16;                    // tile_dim0 = n8 (single line)
  g1[4] = 0u;                                      // tile_dim1=0 (unused), tile_dim2=0
  g1[5] = n8;                                      // tensor_dim0_stride[31:0]
  g1[6] = 0u;                                      // stride0[47:32]=0, stride1[15:0]=0
  g1[7] = 0u;                                      // tensor_dim1_stride[47:16]=0
  // 2-operand form: VADDR2/VADDR3 = NULL (tensor <= 2D), not in a cluster.
  asm volatile("tensor_load_to_lds %0, %1" :: "s"(g0), "s"(g1) : "memory");
}

__device__ __forceinline__ float blockSum(float v, volatile float* red, int tid) {
#pragma unroll
  for (int o = 16; o > 0; o >>= 1) v += __shfl_xor(v, o, 32);
  const int wid = tid >> 5, lane = tid & 31;
  __syncthreads();                 // protect red[] from previous use
  if (lane == 0) red[wid] = v;
  __syncthreads();
  if (wid == 0) {
    float x = red[lane];           // NWAVES == 32 exactly
#pragma unroll
    for (int o = 16; o > 0; o >>= 1) x += __shfl_xor(x, o, 32);
    if (lane == 0) red[0] = x;
  }
  __syncthreads();
  return red[0];
}

__device__ __forceinline__ float blockMax(float v, volatile float* red, int tid) {
#pragma unroll
  for (int o = 16; o > 0; o >>= 1) v = fmaxf(v, __shfl_xor(v, o, 32));
  const int wid = tid >> 5, lane = tid & 31;
  __syncthreads();
  if (lane == 0) red[wid] = v;
  __syncthreads();
  if (wid == 0) {
    float x = red[lane];
#pragma unroll
    for (int o = 16; o > 0; o >>= 1) x = fmaxf(x, __shfl_xor(x, o, 32));
    if (lane == 0) red[0] = x;
  }
  __syncthreads();
  return red[0];
}

__global__ __launch_bounds__(NT) void akl_row_kernel(
    const float* __restrict__ tlog, const float* __restrict__ slog,
    const int* __restrict__ mask, float* __restrict__ perpos, int V) {
  extern __shared__ __align__(16) float lds[];   // raw teacher row, raw student row
  float* zt = lds;
  float* zs = lds + V;
  __shared__ float hist[NB];
  __shared__ float sfx[NB];
  __shared__ float red[NWAVES];
  __shared__ float bcast;
  __shared__ int   selBin;
  __shared__ int   candCount;
  __shared__ float cand[CAP];

  const int row = blockIdx.x;
  const int tid = threadIdx.x;
  const int wid = tid >> 5;
  const float* tr = tlog + (size_t)row * V;
  const float* sr = slog + (size_t)row * V;

  // ---- Phase A: stage both rows into LDS via the Tensor Data Mover.
  // Wave 0 and wave 1 issue independent descriptors (concurrent DMA), each
  // waits its own TENSORcnt; the workgroup barrier releases everyone else.
  const unsigned rowBytes = (unsigned)V << 2;
  const bool tdmOK = ((rowBytes & 7u) == 0u) && ((rowBytes >> 3) <= 0xFFFFu);
  if (tdmOK) {
    if (wid == 0) {
      tdm_load_row(tr, zt, rowBytes);
      __builtin_amdgcn_s_wait_tensorcnt(0);
    } else if (wid == 1) {
      tdm_load_row(sr, zs, rowBytes);
      __builtin_amdgcn_s_wait_tensorcnt(0);
    }
  } else {                                        // generic fallback copy
    const int V4 = V & ~3;
    for (int i = tid * 4; i < V4; i += NT * 4) {
      *(float4*)(zt + i) = *(const float4*)(tr + i);
      *(float4*)(zs + i) = *(const float4*)(sr + i);
    }
    for (int i = V4 + tid; i < V; i += NT) { zt[i] = tr[i]; zs[i] = sr[i]; }
  }
  __syncthreads();

  // ---- Phase A2: row max/min from LDS (z = raw * 0.5f is exact, applied at use)
  float tmax = -__builtin_inff(), smax = -__builtin_inff(), tmin = __builtin_inff();
  for (int i = tid; i < V; i += NT) {
    const float a = zt[i], b = zs[i];
    tmax = fmaxf(tmax, a); tmin = fminf(tmin, a); smax = fmaxf(smax, b);
  }
  const float Tmax = INV_T * blockMax(tmax, red, tid);
  const float Smax = INV_T * blockMax(smax, red, tid);
  const float Tmin = INV_T * (-blockMax(-tmin, red, tid));

  // ---- Phase B: log-sum-exp of both rows (LDS resident)
  float st = 0.f, ss = 0.f;
  for (int i = tid; i < V; i += NT) {
    st += __expf(zt[i] * INV_T - Tmax);
    ss += __expf(zs[i] * INV_T - Smax);
  }
  st = blockSum(st, red, tid);
  ss = blockSum(ss, red, tid);
  const float lseT = Tmax + __logf(st);
  const float lseS = Smax + __logf(ss);

  // ---- Phase C: 2-level histogram select of cutoff logit z*
  // invariant: z* in [curLo, curHi), massAbove = teacher mass with z >= curHi
  float curLo = Tmin;
  float curHi = __builtin_inff();
  const float topC = Tmax + fmaxf(fabsf(Tmax) * 1e-6f, 1e-20f); // strictly > Tmax
  float massAbove = 0.f;

  for (int iter = 0; iter < 2; ++iter) {
    hist[tid] = 0.f;                                  // NB == NT
    if (tid == 0) selBin = -1;
    __syncthreads();
    const float hiC  = (curHi == __builtin_inff()) ? topC : curHi;
    const float w    = (hiC - curLo) * (1.0f / NB);
    const float invw = 1.0f / w;
    for (int i = tid; i < V; i += NT) {
      const float z = zt[i] * INV_T;
      if (z >= curLo && z < curHi) {
        int b = (int)((z - curLo) * invw);
        b = min(max(b, 0), NB - 1);
        atomicAdd(&hist[b], __expf(z - lseT));        // ds_add_f32
      }
    }
    __syncthreads();
    // suffix sum sfx[b] = sum_{j>=b} hist[j]  (Hillis-Steele, 1 bin/thread)
    sfx[tid] = hist[tid];
    __syncthreads();
    for (int off = 1; off < NB; off <<= 1) {
      const float add = (tid + off < NB) ? sfx[tid + off] : 0.f;
      __syncthreads();
      sfx[tid] += add;
      __syncthreads();
    }
    // topmost bin b with massAbove + sfx[b] >= MU  (crossing is unique)
    const float cb    = massAbove + sfx[tid];
    const float cnext = (tid + 1 < NB) ? (massAbove + sfx[tid + 1]) : massAbove;
    if (cb >= MU && (tid == NB - 1 || cnext < MU)) atomicMax(&selBin, tid);
    __syncthreads();
    int bsel = selBin;
    if (bsel < 0) bsel = 0;                           // fp-drift fallback
    const float nMassAbove = massAbove + ((bsel + 1 < NB) ? sfx[bsel + 1] : 0.f);
    const float nLo = curLo + (float)bsel * w;
    const float nHi = curLo + (float)(bsel + 1) * w;
    __syncthreads();                                  // before hist reuse
    curLo = nLo; curHi = nHi; massAbove = nMassAbove;
  }

  // ---- Phase D: gather the few candidates in [curLo, curHi), exact select
  if (tid == 0) candCount = 0;
  __syncthreads();
  for (int i = tid; i < V; i += NT) {
    const float z = zt[i] * INV_T;
    if (z >= curLo && z < curHi) {
      const int idx = atomicAdd(&candCount, 1);
      if (idx < CAP) cand[idx] = z;
    }
  }
  __syncthreads();
  if (tid == 0) {
    const int n = candCount;
    float zc = curLo;                                 // conservative fallback
    if (n > 0 && n <= CAP) {
      float cum = massAbove;
      float used = __builtin_inff();                  // pick strictly below this
      for (int k = 0; k < n; ++k) {
        float m = -__builtin_inff();
        for (int j = 0; j < n; ++j) {
          const float c = cand[j];
          if (c < used && c > m) m = c;
        }
        if (m == -__builtin_inff()) break;
        const float pm = __expf(m - lseT);
        for (int j = 0; j < n; ++j)                   // tie-aware mass
          if (cand[j] == m) cum += pm;
        if (cum >= MU) { zc = m; break; }
        used = m;
      }
    }
    bcast = zc;
  }
  __syncthreads();
  const float zcut = bcast;

  // ---- Phase E: fused fkl / rkl / g_head / g_tail reductions
  float fkl = 0.f, rkl = 0.f, gh = 0.f, gt = 0.f;
  for (int i = tid; i < V; i += NT) {
    const float zi  = zt[i] * INV_T;
    const float ztl = zi - lseT;
    const float zsl = zs[i] * INV_T - lseS;
    const float tp = __expf(ztl), sp = __expf(zsl);
    const float d = ztl - zsl;
    fkl += tp * d;
    rkl -= sp * d;
    const float ad = fabsf(tp - sp);
    if (zi >= zcut) gh += ad; else gt += ad;
  }
  fkl = blockSum(fkl, red, tid);
  rkl = blockSum(rkl, red, tid);
  gh  = blockSum(gh,  red, tid);
  gt  = blockSum(gt,  red, tid);
  if (tid == 0) {
    const float alpha = gh / (gh + gt + AKL_EPS);
    const float pp = alpha * fkl + (1.f - alpha) * rkl;
    perpos[row] = pp * (float)mask[row];
  }
}

__global__ __launch_bounds__(NT) void akl_finalize_kernel(
    const float* __restrict__ perpos, const int* __restrict__ mask,
    float* __restrict__ out, int rows) {
  __shared__ float red[NWAVES];
  const int tid = threadIdx.x;
  float s = 0.f, c = 0.f;
  for (int i = tid; i < rows; i += NT) {
    s += perpos[i];
    c += (float)mask[i];
  }
  s = blockSum(s, red, tid);
  c = blockSum(c, red, tid);
  if (tid == 0) out[0] = s / fmaxf(c, 1.0f) * T2;
}

extern "C" void kernel_launch(void* const* d_in, const int* in_sizes, int n_in,
                              void* d_out, int out_size, void* d_ws, size_t ws_size,
                              hipStream_t stream) {
  const float* teacher = (const float*)d_in[0];
  const float* student = (const float*)d_in[1];
  const int*   mask    = (const int*)d_in[2];
  const int rows = in_sizes[2];                 // B*L
  const int V    = in_sizes[0] / rows;          // vocab
  float* perpos  = (float*)d_ws;                // rows floats of scratch

  const size_t shmem = (size_t)2 * (size_t)V * sizeof(float);  // ~250 KB < 320 KB WGP LDS
  (void)hipFuncSetAttribute((const void*)akl_row_kernel,
                            hipFuncAttributeMaxDynamicSharedMemorySize, (int)shmem);

  akl_row_kernel<<<rows, NT, shmem, stream>>>(teacher, student, mask, perpos, V);
  akl_finalize_kernel<<<1, NT, 0, stream>>>(perpos, mask, (float*)d_out, rows);
}